// RtStgcnLayer_18760417148962
// MI455X (gfx1250) — compile-verified
//
#include <hip/hip_runtime.h>
#include <cstddef>

// ---------------- problem constants ----------------
#define N_    8
#define CIN   256
#define COUT  256
#define L_    512
#define V_    25
#define P_    3
#define KTAP  9
// ---------------- tiling ----------------
#define CBLK  16          // channels of C_OUT per workgroup
#define LBLK  8           // l values per workgroup
#define KCH   16          // K-chunk (C_IN) staged in LDS
#define VPAD  32          // v padded to 32 -> clean 16-wide N tiles
#define COLS  (LBLK*VPAD) // 256 columns -> 16 N-tiles
#define NTW   2           // N-tiles per wave (16 tiles / 8 waves)
// ---------------- LDS layout (floats) ----------------
#define XL_OFF 0
#define XL_SZ  (KCH*LBLK*VPAD)            // 4096
#define WL_OFF XL_SZ                       // W tile: [48][17] (pad vs banks)
#define WL_SZ  (48*17)                     // 816
#define YSTR   26
#define YL_OFF 0                           // reuses X/W region after GEMM
#define YL_SZ  (P_*CBLK*LBLK*YSTR)         // 9984
#define AL_OFF YL_SZ                       // A matrices, loaded once
#define AL_SZ  (P_*V_*V_)                  // 1875
#define SMEM_FLOATS (AL_OFF + AL_SZ)       // 11859 floats ~ 47.4 KB

typedef float v2f __attribute__((ext_vector_type(2)));
typedef float v8f __attribute__((ext_vector_type(8)));

__global__ __launch_bounds__(256)
void stgcn_gemm_mix_kernel(const float* __restrict__ x,
                           const float* __restrict__ conv_w,
                           const float* __restrict__ Amat,
                           float* __restrict__ s)
{
    __shared__ float smem[SMEM_FLOATS];

    const int tid  = threadIdx.x;
    const int wave = tid >> 5;
    const int lane = tid & 31;
    const int m    = lane & 15;    // row/col within 16
    const int hi   = lane >> 4;    // lane half

    const int bid = blockIdx.x;          // 8192 = 8 * 16 * 64
    const int n   = bid >> 10;
    const int rem = bid & 1023;
    const int ct  = rem >> 6;
    const int lt  = rem & 63;
    const int c0  = ct * CBLK;
    const int l0  = lt * LBLK;

    // stage A (graph adjacency) once; region disjoint from GEMM scratch
    for (int i = tid; i < AL_SZ; i += 256) smem[AL_OFF + i] = Amat[i];

    // fp32 accumulators: 3 partitions x 2 N-tiles, 8 VGPRs each
    v8f acc[P_][NTW];
#pragma unroll
    for (int p = 0; p < P_; ++p)
#pragma unroll
        for (int j = 0; j < NTW; ++j)
#pragma unroll
            for (int r = 0; r < 8; ++r) acc[p][j][r] = 0.0f;

    for (int kc = 0; kc < CIN / KCH; ++kc) {
        const int k0 = kc * KCH;

        // ---- stage X chunk: [kk][l][vpad32], zero pad v>=25 ----
        for (int i = tid; i < XL_SZ; i += 256) {
            const int kk = i >> 8;          // / (LBLK*VPAD)
            const int r2 = i & 255;
            const int l  = r2 >> 5;
            const int v  = r2 & 31;
            float val = 0.0f;
            if (v < V_)
                val = x[(((size_t)n * CIN + (k0 + kk)) * L_ + (l0 + l)) * V_ + v];
            smem[XL_OFF + i] = val;
        }
        // ---- stage W chunk: 48 rows (3p x 16c) x 16 k, stride 17 ----
        for (int i = tid; i < 48 * KCH; i += 256) {
            const int r  = i >> 4;
            const int kk = i & 15;
            const int p  = r >> 4;
            const int c  = r & 15;
            smem[WL_OFF + r * 17 + kk] =
                conv_w[((size_t)(p * COUT + c0 + c)) * CIN + (k0 + kk)];
        }
        // hint next chunk toward L2 while we compute this one
        if (kc + 1 < CIN / KCH)
            __builtin_prefetch(
                &x[(((size_t)n * CIN + (k0 + KCH + (tid >> 4))) * L_ + l0) * V_], 0, 0);
        __syncthreads();

        // ---- 4 WMMA k-steps of K=4 over this 16-wide chunk ----
#pragma unroll
        for (int ks = 0; ks < 4; ++ks) {
            const int kb = ks * 4 + 2 * hi;  // ISA A-layout: lanes>=16 hold K+2
            v2f a[P_];
#pragma unroll
            for (int p = 0; p < P_; ++p) {
                a[p].x = smem[WL_OFF + (p * 16 + m) * 17 + kb];
                a[p].y = smem[WL_OFF + (p * 16 + m) * 17 + kb + 1];
            }
            v2f b[NTW];
#pragma unroll
            for (int j = 0; j < NTW; ++j) {
                const int col = (wave + 8 * j) * 16 + m;
                const int l   = col >> 5;
                const int vv  = col & 31;
                b[j].x = smem[XL_OFF + ((kb + 0) * LBLK + l) * VPAD + vv];
                b[j].y = smem[XL_OFF + ((kb + 1) * LBLK + l) * VPAD + vv];
            }
#pragma unroll
            for (int p = 0; p < P_; ++p)
#pragma unroll
                for (int j = 0; j < NTW; ++j)
                    acc[p][j] = __builtin_amdgcn_wmma_f32_16x16x4_f32(
                        false, a[p], false, b[j], (short)0, acc[p][j],
                        false, false);
        }
        __syncthreads();
    }

    // ---- dump Y tiles to LDS: [p][c16][l8][26] (vv>=25 are exact zeros) ----
#pragma unroll
    for (int p = 0; p < P_; ++p)
#pragma unroll
        for (int j = 0; j < NTW; ++j) {
            const int col = (wave + 8 * j) * 16 + m;
            const int l   = col >> 5;
            const int vv  = col & 31;
            if (vv < YSTR) {
#pragma unroll
                for (int r = 0; r < 8; ++r) {
                    const int c = r + 8 * hi;   // ISA C/D layout
                    smem[YL_OFF + ((p * CBLK + c) * LBLK + l) * YSTR + vv] =
                        acc[p][j][r];
                }
            }
        }
    __syncthreads();

    // ---- graph mix: s[c,l,w] = sum_p sum_v Y[p,c,l,v] * A[p,v,w] ----
    // 256 threads = 16c x 8l x 2 w-phases
    const int c  = tid >> 4;
    const int l  = (tid >> 1) & 7;
    const int wh = tid & 1;

    float yreg[P_ * V_];
#pragma unroll
    for (int p = 0; p < P_; ++p)
#pragma unroll
        for (int v = 0; v < V_; ++v)
            yreg[p * V_ + v] =
                smem[YL_OFF + ((p * CBLK + c) * LBLK + l) * YSTR + v];

    for (int w = wh; w < V_; w += 2) {
        float av = 0.0f;
#pragma unroll
        for (int p = 0; p < P_; ++p)
#pragma unroll
            for (int v = 0; v < V_; ++v)
                av += yreg[p * V_ + v] * smem[AL_OFF + (p * V_ + v) * V_ + w];
        s[(((size_t)n * COUT + (c0 + c)) * L_ + (l0 + l)) * V_ + w] = av;
    }
}

// ---- temporal 9-tap sum + BN + ReLU + residual + ReLU (L2-resident s) ----
__global__ __launch_bounds__(256)
void stgcn_tail_kernel(const float* __restrict__ s,
                       const float* __restrict__ x,
                       const float* __restrict__ gamma,
                       const float* __restrict__ beta,
                       const float* __restrict__ mean,
                       const float* __restrict__ var,
                       float* __restrict__ out)
{
    const size_t idx = (size_t)blockIdx.x * 256 + threadIdx.x;
    const int w = (int)(idx % V_);
    size_t t    = idx / V_;
    const int l = (int)(t % L_);
    t /= L_;
    const int c = (int)(t % COUT);
    const int n = (int)(t / COUT);

    const float* sp = s + (((size_t)n * COUT + c) * L_) * V_ + w;
    float av = 0.0f;
#pragma unroll
    for (int k = 0; k < KTAP; ++k) {
        const int li = l - (KTAP - 1) + k;
        if (li >= 0) av += sp[(size_t)li * V_];
    }
    const float sc = gamma[c] * rsqrtf(var[c] + 1e-5f);
    const float sh = beta[c] - mean[c] * sc;
    float r = fmaxf(av * sc + sh, 0.0f);
    r = fmaxf(r + x[idx], 0.0f);
    out[idx] = r;
}

extern "C" void kernel_launch(void* const* d_in, const int* in_sizes, int n_in,
                              void* d_out, int out_size, void* d_ws, size_t ws_size,
                              hipStream_t stream) {
    (void)in_sizes; (void)n_in; (void)out_size; (void)ws_size;
    const float* x      = (const float*)d_in[0];
    const float* conv_w = (const float*)d_in[1];
    const float* Amat   = (const float*)d_in[2];
    const float* gamma  = (const float*)d_in[3];
    const float* beta   = (const float*)d_in[4];
    const float* mean   = (const float*)d_in[5];
    const float* var    = (const float*)d_in[6];
    float* out = (float*)d_out;
    float* s   = (float*)d_ws;   // 8*256*512*25*4 = 104.86 MB scratch

    const int g1 = N_ * (COUT / CBLK) * (L_ / LBLK);        // 8192 WGs
    stgcn_gemm_mix_kernel<<<g1, 256, 0, stream>>>(x, conv_w, Amat, s);

    const int total = N_ * COUT * L_ * V_;                  // 26,214,400
    stgcn_tail_kernel<<<total / 256, 256, 0, stream>>>(s, x, gamma, beta,
                                                       mean, var, out);
}